// RPEMultiHeadAttention_56831007261094
// MI455X (gfx1250) — compile-verified
//
#include <hip/hip_runtime.h>
#include <hip/hip_bf16.h>
#include <math.h>

// Problem constants (from reference): B=2, N=512, D=256, H=4, d=64
#define NB 2
#define NS 512
#define DM 256
#define NH 4
#define HD 64

typedef float v2f __attribute__((ext_vector_type(2)));
typedef float v8f __attribute__((ext_vector_type(8)));

// D = A(16x4,f32) * B(4x16,f32) + C(16x16,f32)
__device__ __forceinline__ v8f wmma4(v2f a, v2f b, v8f c) {
  return __builtin_amdgcn_wmma_f32_16x16x4_f32(
      /*neg_a=*/false, a, /*neg_b=*/false, b,
      /*c_mod=*/(short)0, c, /*reuse_a=*/false, /*reuse_b=*/false);
}

__device__ __forceinline__ int waveId() {
  return blockIdx.x * (blockDim.x >> 5) + (threadIdx.x >> 5);
}

// ---------------------------------------------------------------------------
// Kernel 1: Q/K/V projections.  out = x(1024x256) @ W.T + bias
// One wave per 16x16 output tile. waves = 3 * (1024/16) * (256/16) = 3072
// ---------------------------------------------------------------------------
__global__ void qkv_proj(const float* __restrict__ x,
                         const float* __restrict__ Wq, const float* __restrict__ bq,
                         const float* __restrict__ Wk, const float* __restrict__ bk,
                         const float* __restrict__ Wv, const float* __restrict__ bv,
                         float* __restrict__ Q, float* __restrict__ K,
                         float* __restrict__ V) {
  int w = waveId();
  int which = w / (64 * 16);
  int rem   = w % (64 * 16);
  int rt = rem / 16, ct = rem % 16;
  const float* W    = (which == 0) ? Wq : (which == 1) ? Wk : Wv;
  const float* bias = (which == 0) ? bq : (which == 1) ? bk : bv;
  float*       out  = (which == 0) ? Q  : (which == 1) ? K  : V;

  int lane = threadIdx.x & 31, half = lane >> 4, l = lane & 15;
  int row0 = rt * 16, col0 = ct * 16;
  v8f c = {};
  for (int k0 = 0; k0 < DM; k0 += 4) {
    // A[row][k] = x[row0+l][k0+2*half .. +1]
    v2f a = *(const v2f*)(x + (size_t)(row0 + l) * DM + k0 + 2 * half);
    // B[k][col] = W.T[k][col] = W[col0+l][k0+2*half .. +1]  (contiguous)
    v2f b = *(const v2f*)(W + (size_t)(col0 + l) * DM + k0 + 2 * half);
    c = wmma4(a, b, c);
  }
  float bv_ = bias[col0 + l];
  for (int r = 0; r < 8; ++r) {
    int row = row0 + half * 8 + r;
    out[(size_t)row * DM + col0 + l] = c[r] + bv_;
  }
}

// ---------------------------------------------------------------------------
// Kernel 2: QW[b,h,n,i] = sum_d' Q[b,n,h*64+d'] * Wr[h*64+d'][i]
// Per (b,h): A = Q_head (512x64), B = Wr rows h*64..h*64+63 (64x256).
// waves = NB*NH*(NS/16)*(DM/16) = 2*4*32*16 = 4096
// ---------------------------------------------------------------------------
__global__ void qw_proj(const float* __restrict__ Q, const float* __restrict__ Wr,
                        float* __restrict__ QW) {
  int w = waveId();
  int ct = w % 16;  w /= 16;
  int rt = w % 32;  w /= 32;
  int h  = w % NH;
  int b  = w / NH;

  int lane = threadIdx.x & 31, half = lane >> 4, l = lane & 15;
  int n0 = rt * 16, col0 = ct * 16;
  v8f c = {};
  for (int k0 = 0; k0 < HD; k0 += 4) {
    v2f a = *(const v2f*)(Q + ((size_t)b * NS + n0 + l) * DM + h * HD + k0 + 2 * half);
    int krow = h * HD + k0 + 2 * half;  // B[k][col] = Wr[krow][col0+l]
    v2f bb;
    bb[0] = Wr[(size_t)krow * DM + col0 + l];
    bb[1] = Wr[(size_t)(krow + 1) * DM + col0 + l];
    c = wmma4(a, bb, c);
  }
  for (int r = 0; r < 8; ++r) {
    int n = n0 + half * 8 + r;
    QW[(((size_t)b * NH + h) * NS + n) * DM + col0 + l] = c[r];
  }
}

// ---------------------------------------------------------------------------
// Kernel 3: content logits  S[b,h,n,m] = Q_h[n,:] . K_h[m,:]
// waves = NB*NH*32*32 = 8192
// ---------------------------------------------------------------------------
__global__ void qk_content(const float* __restrict__ Q, const float* __restrict__ K,
                           float* __restrict__ S) {
  int w = waveId();
  int mt = w % 32;  w /= 32;
  int nt = w % 32;  w /= 32;
  int h  = w % NH;
  int b  = w / NH;

  int lane = threadIdx.x & 31, half = lane >> 4, l = lane & 15;
  int n0 = nt * 16, m0 = mt * 16;
  v8f c = {};
  for (int k0 = 0; k0 < HD; k0 += 4) {
    v2f a  = *(const v2f*)(Q + ((size_t)b * NS + n0 + l) * DM + h * HD + k0 + 2 * half);
    // B[k][col] = K[b, m0+col, h*64+k] -> per lane two consecutive k
    v2f bb = *(const v2f*)(K + ((size_t)b * NS + m0 + l) * DM + h * HD + k0 + 2 * half);
    c = wmma4(a, bb, c);
  }
  for (int r = 0; r < 8; ++r) {
    int n = n0 + half * 8 + r;
    S[(((size_t)b * NH + h) * NS + n) * NS + m0 + l] = c[r];
  }
}

// ---------------------------------------------------------------------------
// Kernel 4 (HBM-bound, streams 512MB of rpe exactly once):
//   S[b,h,n,m] += rpe[b,n,m,:] . QW[b,h,n,:]
// One wave per (b, n, group of 4 m-tiles).  GEMM per tile: A = rpe rows
// (16m x 256, NON-TEMPORAL loads: read-once stream must not evict QW/S
// from the 192MB L2), B = QW[b,:,n,:]^T (256 x 16, cols 0..3 = heads,
// rest zero).  One B-fragment load feeds 4 WMMAs (4 m-tiles) per k-step.
// Content logits enter through the WMMA C operand.
// waves = NB*NS*8 = 8192
// ---------------------------------------------------------------------------
__global__ void rpe_pos(const float* __restrict__ rpe, const float* __restrict__ QW,
                        float* __restrict__ S) {
  int w = waveId();
  int g = w % 8;   w /= 8;       // group of 4 m-tiles: m in [g*64, g*64+64)
  int n = w % NS;
  int b = w / NS;

  int lane = threadIdx.x & 31, half = lane >> 4, l = lane & 15;
  int mbase = g * 64;
  const float* rbase = rpe + (((size_t)b * NS + n) * NS) * DM;

  v8f c[4];
  #pragma unroll
  for (int j = 0; j < 4; ++j) {
    v8f z = {};
    c[j] = z;
  }
  if (l < NH) {  // load existing content logits as accumulator (cols 4..15 dead)
    const float* srow = S + (((size_t)b * NH + l) * NS + n) * NS + mbase + half * 8;
    #pragma unroll
    for (int j = 0; j < 4; ++j)
      #pragma unroll
      for (int r = 0; r < 8; ++r) c[j][r] = srow[j * 16 + r];
  }

  const float* qwrow = QW + (((size_t)b * NH + (l < NH ? l : 0)) * NS + n) * DM;
  bool live = (l < NH);
  for (int k0 = 0; k0 < DM; k0 += 4) {
    // B[k][col] = QW[b,col,n,k] (cols >= NH are zero); shared by all 4 tiles
    v2f bb = *(const v2f*)(qwrow + k0 + 2 * half);
    if (!live) { bb[0] = 0.f; bb[1] = 0.f; }
    #pragma unroll
    for (int j = 0; j < 4; ++j) {
      // A[row][k] = rpe[b,n, mbase+16j+l, k0+2*half .. +1]  (streaming, NT)
      v2f a = __builtin_nontemporal_load(
          (const v2f*)(rbase + (size_t)(mbase + j * 16 + l) * DM + k0 + 2 * half));
      c[j] = wmma4(a, bb, c[j]);
    }
  }

  if (l < NH) {
    float* srow = S + (((size_t)b * NH + l) * NS + n) * NS + mbase + half * 8;
    #pragma unroll
    for (int j = 0; j < 4; ++j)
      #pragma unroll
      for (int r = 0; r < 8; ++r) srow[j * 16 + r] = c[j][r];
  }
}

// ---------------------------------------------------------------------------
// Kernel 5: masked, scaled softmax over m.  One wave per (b,h,n) row.
// waves = NB*NH*NS = 4096
// ---------------------------------------------------------------------------
__global__ void softmax_rows(float* __restrict__ S, const unsigned char* __restrict__ mask) {
  int w = waveId();
  int n = w % NS;  w /= NS;
  int h = w % NH;
  int b = w / NH;
  int lane = threadIdx.x & 31;

  float* row = S + (((size_t)b * NH + h) * NS + n) * NS;
  const unsigned char* mrow = mask + ((size_t)b * NS + n) * NS;
  const float scale = 0.125f;  // 1/sqrt(64)

  float vals[16];
  float mx = -INFINITY;
  for (int i = 0; i < 16; ++i) {
    int m = i * 32 + lane;
    float v = mrow[m] ? row[m] * scale : -INFINITY;
    vals[i] = v;
    mx = fmaxf(mx, v);
  }
  for (int off = 16; off > 0; off >>= 1)
    mx = fmaxf(mx, __shfl_xor(mx, off, 32));

  float sum = 0.f;
  for (int i = 0; i < 16; ++i) {
    float e = (vals[i] == -INFINITY) ? 0.f : __expf(vals[i] - mx);
    vals[i] = e;
    sum += e;
  }
  for (int off = 16; off > 0; off >>= 1)
    sum += __shfl_xor(sum, off, 32);
  float inv = (sum > 0.f) ? 1.f / sum : 0.f;

  for (int i = 0; i < 16; ++i)
    row[i * 32 + lane] = vals[i] * inv;
}

// ---------------------------------------------------------------------------
// Kernel 6: O[b,n,h*64+dd] = sum_m P[b,h,n,m] * V[b,m,h*64+dd]
// waves = NB*NH*32*(64/16) = 1024
// ---------------------------------------------------------------------------
__global__ void attn_v(const float* __restrict__ S, const float* __restrict__ V,
                       float* __restrict__ O) {
  int w = waveId();
  int dt = w % 4;   w /= 4;
  int nt = w % 32;  w /= 32;
  int h  = w % NH;
  int b  = w / NH;

  int lane = threadIdx.x & 31, half = lane >> 4, l = lane & 15;
  int n0 = nt * 16, d0 = dt * 16;
  v8f c = {};
  for (int k0 = 0; k0 < NS; k0 += 4) {
    v2f a = *(const v2f*)(S + (((size_t)b * NH + h) * NS + n0 + l) * NS + k0 + 2 * half);
    int kk = k0 + 2 * half;  // B[k][col] = V[b, k, h*64+d0+col]
    v2f bb;
    bb[0] = V[((size_t)b * NS + kk)     * DM + h * HD + d0 + l];
    bb[1] = V[((size_t)b * NS + kk + 1) * DM + h * HD + d0 + l];
    c = wmma4(a, bb, c);
  }
  for (int r = 0; r < 8; ++r) {
    int n = n0 + half * 8 + r;
    O[((size_t)b * NS + n) * DM + h * HD + d0 + l] = c[r];
  }
}

// ---------------------------------------------------------------------------
// Kernel 7: out = O(1024x256) @ Wo.T + bo.  waves = 64*16 = 1024
// ---------------------------------------------------------------------------
__global__ void out_proj(const float* __restrict__ O, const float* __restrict__ Wo,
                         const float* __restrict__ bo, float* __restrict__ out) {
  int w = waveId();
  int rt = w / 16, ct = w % 16;
  int lane = threadIdx.x & 31, half = lane >> 4, l = lane & 15;
  int row0 = rt * 16, col0 = ct * 16;
  v8f c = {};
  for (int k0 = 0; k0 < DM; k0 += 4) {
    v2f a = *(const v2f*)(O  + (size_t)(row0 + l) * DM + k0 + 2 * half);
    v2f b = *(const v2f*)(Wo + (size_t)(col0 + l) * DM + k0 + 2 * half);
    c = wmma4(a, b, c);
  }
  float bv_ = bo[col0 + l];
  for (int r = 0; r < 8; ++r) {
    int row = row0 + half * 8 + r;
    out[(size_t)row * DM + col0 + l] = c[r] + bv_;
  }
}

// ---------------------------------------------------------------------------
extern "C" void kernel_launch(void* const* d_in, const int* in_sizes, int n_in,
                              void* d_out, int out_size, void* d_ws, size_t ws_size,
                              hipStream_t stream) {
  const float* x    = (const float*)d_in[0];
  const float* rpe  = (const float*)d_in[1];
  const unsigned char* mask = (const unsigned char*)d_in[2];
  const float* Wq = (const float*)d_in[3];
  const float* bq = (const float*)d_in[4];
  const float* Wk = (const float*)d_in[5];
  const float* bk = (const float*)d_in[6];
  const float* Wv = (const float*)d_in[7];
  const float* bv = (const float*)d_in[8];
  const float* Wr = (const float*)d_in[9];
  const float* Wo = (const float*)d_in[10];
  const float* bo = (const float*)d_in[11];
  float* out = (float*)d_out;

  // Workspace layout (floats), ~16 MB total
  float* ws = (float*)d_ws;
  const size_t szX = (size_t)NB * NS * DM;           // 262144
  float* Q  = ws;
  float* K  = Q  + szX;
  float* V  = K  + szX;
  float* QW = V  + szX;                              // NB*NH*NS*DM = 1M floats
  float* S  = QW + (size_t)NB * NH * NS * DM;        // NB*NH*NS*NS = 2M floats
  float* O  = S  + (size_t)NB * NH * NS * NS;

  const int BLK = 256;              // 8 waves per workgroup
  const int WPB = BLK / 32;

  qkv_proj    <<<(3 * 64 * 16) / WPB, BLK, 0, stream>>>(x, Wq, bq, Wk, bk, Wv, bv, Q, K, V);
  qw_proj     <<<(NB * NH * 32 * 16) / WPB, BLK, 0, stream>>>(Q, Wr, QW);
  qk_content  <<<(NB * NH * 32 * 32) / WPB, BLK, 0, stream>>>(Q, K, S);
  rpe_pos     <<<(NB * NS * 8) / WPB, BLK, 0, stream>>>(rpe, QW, S);
  softmax_rows<<<(NB * NH * NS) / WPB, BLK, 0, stream>>>(S, mask);
  attn_v      <<<(NB * NH * 32 * 4) / WPB, BLK, 0, stream>>>(S, V, O);
  out_proj    <<<(64 * 16) / WPB, BLK, 0, stream>>>(O, Wo, bo, out);
}